// RACNN_28166395527345
// MI455X (gfx1250) — compile-verified
//
#include <hip/hip_runtime.h>
#include <cstdint>
#include <cstddef>

#define IMG_S 224

// h(z) = sigmoid(10 z)
__device__ __forceinline__ float sig10(float z) {
    return 1.0f / (1.0f + expf(-10.0f * z));
}

// Async global -> LDS staging (gfx1250 ASYNCcnt path).
// %0 = LDS byte address (VDST field), %1 = 64-bit global address, no SADDR.
__device__ __forceinline__ void async_ld_b128(uint32_t lds_off, const float* gaddr) {
    asm volatile("global_load_async_to_lds_b128 %0, %1, off"
                 :: "v"(lds_off), "v"(gaddr) : "memory");
}

__device__ __forceinline__ void wait_asynccnt0() {
#if __has_builtin(__builtin_amdgcn_s_wait_asynccnt)
    __builtin_amdgcn_s_wait_asynccnt(0);
#else
    asm volatile("s_wait_asynccnt 0" ::: "memory");
#endif
}

__global__ __launch_bounds__(IMG_S)
void racnn_attention_crop(const float* __restrict__ apn,
                          const float* __restrict__ in,
                          float* __restrict__ out) {
    const int u = blockIdx.x;   // output row
    const int b = blockIdx.y;   // batch image
    const int v = threadIdx.x;  // output column (0..223)

    // LDS staging buffer: [ch*2 + {r0,r1}][224 columns]
    __shared__ __align__(16) float rows[6 * IMG_S];

    // ---- box parameters (uniform across block; SALU-friendly) ----
    const float a0 = apn[b * 3 + 0];
    const float a1 = apn[b * 3 + 1];
    const float a2 = apn[b * 3 + 2];
    const float tx = 112.0f + truncf(a0 * 56.0f + 0.5f);
    const float ty = 112.0f + truncf(a1 * 56.0f + 0.5f);
    const float tl = 38.0f + truncf((a2 + 1.0f) * 0.5f * 18.0f);
    const int   r_hi = (int)(tx + tl - 1.0f);
    const int   c_hi = (int)(ty + tl - 1.0f);
    // keep exact reference rounding: (2*tl)/224, then scaled per sample
    const float scale   = (2.0f * tl) / (float)IMG_S;
    const float src_max = 2.0f * tl - 1.0f;

    // ---- source row for this output row ----
    float sr = ((float)u + 0.5f) * scale - 0.5f;
    sr = fminf(fmaxf(sr, 0.0f), src_max);
    const float r   = sr + (tx - tl);
    const float r0f = floorf(r);
    const float wr  = r - r0f;
    const int   r0  = (int)r0f;
    const int   r1  = min(r0 + 1, r_hi);

    // ---- issue async staging of rows r0,r1 (3 channels) into LDS ----
    // 6 segments x 224 floats = 336 x 16B transfers, in flight while we
    // compute the per-thread column taps below (latency overlap).
    const float*   base    = in + (size_t)b * 3 * IMG_S * IMG_S;
    const uint32_t ldsBase = (uint32_t)(uintptr_t)(&rows[0]); // low 32 bits = LDS offset
    for (int i = v; i < 6 * (IMG_S / 4); i += IMG_S) {
        const int seg = i / (IMG_S / 4);        // 0..5
        const int off = (i % (IMG_S / 4)) * 4;  // float offset within the row
        const int ch  = seg >> 1;
        const int row = (seg & 1) ? r1 : r0;
        const float* g = base + ((size_t)ch * IMG_S + row) * IMG_S + off;
        async_ld_b128(ldsBase + (uint32_t)i * 16u, g);
    }

    // ---- per-thread column coords + tap weights (overlaps the DMA) ----
    float sc = ((float)v + 0.5f) * scale - 0.5f;
    sc = fminf(fmaxf(sc, 0.0f), src_max);
    const float c   = sc + (ty - tl);
    const float c0f = floorf(c);
    const float wc  = c - c0f;
    const int   c0  = (int)c0f;
    const int   c1  = min(c0 + 1, c_hi);

    const float mr0 = sig10(r0f - tx + tl) - sig10(r0f - tx - tl);
    const float mr1 = sig10((float)r1 - tx + tl) - sig10((float)r1 - tx - tl);
    const float mc0 = sig10(c0f - ty + tl) - sig10(c0f - ty - tl);
    const float mc1 = sig10((float)c1 - ty + tl) - sig10((float)c1 - ty - tl);

    const float w00 = mr0 * mc0 * (1.0f - wr) * (1.0f - wc);
    const float w01 = mr0 * mc1 * (1.0f - wr) * wc;
    const float w10 = mr1 * mc0 * wr * (1.0f - wc);
    const float w11 = mr1 * mc1 * wr * wc;

    // ---- drain DMA, cross-wave handoff, gather from LDS, store ----
    wait_asynccnt0();
    __syncthreads();

    float* obase = out + ((size_t)b * 3 * IMG_S + u) * IMG_S;
    #pragma unroll
    for (int ch = 0; ch < 3; ++ch) {
        const float* t0 = &rows[(ch * 2 + 0) * IMG_S];
        const float* t1 = &rows[(ch * 2 + 1) * IMG_S];
        const float val = t0[c0] * w00 + t0[c1] * w01 +
                          t1[c0] * w10 + t1[c1] * w11;
        obase[(size_t)ch * IMG_S * IMG_S + v] = val;
    }
}

extern "C" void kernel_launch(void* const* d_in, const int* in_sizes, int n_in,
                              void* d_out, int out_size, void* d_ws, size_t ws_size,
                              hipStream_t stream) {
    const float* apn = (const float*)d_in[0];   // [B,3]
    const float* inp = (const float*)d_in[1];   // [B,3,224,224]
    float*       out = (float*)d_out;           // [B,3,224,224]
    const int B = in_sizes[0] / 3;
    dim3 grid(IMG_S, B);
    racnn_attention_crop<<<grid, IMG_S, 0, stream>>>(apn, inp, out);
}